// MOEfusion_86002425135628
// MI455X (gfx1250) — compile-verified
//
#include <hip/hip_runtime.h>
#include <math.h>

#define B_TOK 1024
#define DIM   1024
#define NEXP  16
#define HID   2048
#define OUT   1024

typedef __attribute__((ext_vector_type(2))) float v2f;
typedef __attribute__((ext_vector_type(8))) float v8f;
typedef __attribute__((ext_vector_type(4))) unsigned int u32x4;
typedef __attribute__((ext_vector_type(8))) int i32x8;
typedef __attribute__((ext_vector_type(4))) int i32x4;

__device__ __forceinline__ v8f wmma_f32(v2f a, v2f b, v8f c) {
  // V_WMMA_F32_16X16X4_F32 : D(16x16 f32) = A(16x4 f32) x B(4x16 f32) + C
  return __builtin_amdgcn_wmma_f32_16x16x4_f32(
      /*neg_a=*/false, a, /*neg_b=*/false, b,
      /*c_mod=*/(short)0, c, /*reuse_a=*/false, /*reuse_b=*/false);
}

// ---------------------------------------------------------------------------
// 0) zero routing counters / stats (pair buffer needs no init: every slot is
//    written exactly once by its owning expert tile)
// ---------------------------------------------------------------------------
__global__ void zero_kernel(int* __restrict__ cnt, float* __restrict__ imp,
                            float* __restrict__ lod) {
  const int i = threadIdx.x;
  if (i < NEXP) { cnt[i] = 0; imp[i] = 0.0f; lod[i] = 0.0f; }
}

// ---------------------------------------------------------------------------
// 1) noisy top-k gating + routing (one 256-thread block per token)
// ---------------------------------------------------------------------------
__global__ __launch_bounds__(256)
void gating_kernel(const float* __restrict__ x, const float* __restrict__ wg,
                   const float* __restrict__ wn, const float* __restrict__ noise,
                   int* __restrict__ cnt, float* __restrict__ imp,
                   float* __restrict__ lod, int* __restrict__ list,
                   float* __restrict__ glist, int* __restrict__ rev) {
  const int b = blockIdx.x;
  const int t = threadIdx.x;
  __shared__ float part[32][9];   // padded: 32 columns x 8 chunks
  __shared__ float logit[32];     // [0..15] clean, [16..31] noise pre-softplus

  const int c  = t & 31;
  const int ch = t >> 5;
  const float* xr = x + (size_t)b * DIM;
  const float* w  = (c < 16) ? (wg + c) : (wn + (c - 16));
  float s = 0.0f;
  const int d0 = ch * (DIM / 8);
  for (int d = d0; d < d0 + DIM / 8; ++d) s += xr[d] * w[(size_t)d * NEXP];
  part[c][ch] = s;
  __syncthreads();
  if (t < 32) {
    float a = 0.0f;
    #pragma unroll
    for (int j = 0; j < 8; ++j) a += part[t][j];
    logit[t] = a;
  }
  __syncthreads();

  if (t == 0) {
    float clean[NEXP], stdv[NEXP], noisy[NEXP];
    #pragma unroll
    for (int e = 0; e < NEXP; ++e) {
      clean[e] = logit[e];
      float z  = logit[NEXP + e];
      float sp = (z > 20.0f) ? z : log1pf(expf(z));   // softplus
      stdv[e]  = sp + 0.1f;                            // + noise_epsilon
      noisy[e] = clean[e] + noise[(size_t)b * NEXP + e] * stdv[e];
    }
    // top-5 of 16 by selection
    int topi[5]; float topv[5]; unsigned used = 0;
    for (int i = 0; i < 5; ++i) {
      float best = -INFINITY; int bi = 0;
      for (int e = 0; e < NEXP; ++e)
        if (!((used >> e) & 1u) && noisy[e] > best) { best = noisy[e]; bi = e; }
      used |= 1u << bi; topi[i] = bi; topv[i] = best;
    }
    // softmax over top-4
    const float m = topv[0];
    float g[4], se = 0.0f;
    #pragma unroll
    for (int i = 0; i < 4; ++i) { g[i] = expf(topv[i] - m); se += g[i]; }
    #pragma unroll
    for (int i = 0; i < 4; ++i) g[i] /= se;
    // smooth load estimate (Normal CDF via erf)
    const float thr_in = topv[4], thr_out = topv[3];
    for (int e = 0; e < NEXP; ++e) {
      const bool  is_in = noisy[e] > thr_in;
      const float thr   = is_in ? thr_in : thr_out;
      const float z     = (clean[e] - thr) / stdv[e];
      const float p     = 0.5f * (1.0f + erff(z * 0.70710678118654752f));
      atomicAdd(lod + e, p);
    }
    // importance + routing scatter; slot id b*4+i owns one pair-output row
    #pragma unroll
    for (int i = 0; i < 4; ++i) {
      const int e = topi[i];
      atomicAdd(imp + e, g[i]);
      const int pos = atomicAdd(cnt + e, 1);
      list[e * B_TOK + pos]  = b;
      glist[e * B_TOK + pos] = g[i];
      rev[e * B_TOK + pos]   = b * 4 + i;
    }
  }
}

// ---------------------------------------------------------------------------
// 2) sparse expert FFN: TDM gather of routed rows -> LDS, fp32 WMMA GEMMs with
//    the full H=2048 activation staged in 128KB LDS (needs CDNA5's 320KB/WGP),
//    fused relu / exp / gate epilogue, deterministic pair-slot stores.
//    grid = NEXP * 64 token tiles; block = 256 (8 wave32); ~192KB dynamic LDS
// ---------------------------------------------------------------------------
__global__ __launch_bounds__(256)
void expert_kernel(const float* __restrict__ x,  const float* __restrict__ W1,
                   const float* __restrict__ b1, const float* __restrict__ W2,
                   const float* __restrict__ b2, const int* __restrict__ cnt,
                   const int* __restrict__ list, const float* __restrict__ glist,
                   const int* __restrict__ rev, float* __restrict__ pair_out) {
  const int e    = blockIdx.x >> 6;   // 64 token tiles per expert
  const int tile = blockIdx.x & 63;
  const int n    = cnt[e];
  if (tile * 16 >= n) return;
  const int rows = min(16, n - tile * 16);

  extern __shared__ float smem[];
  float* xs  = smem;                  // [16][DIM]  64 KB
  float* hs  = smem + 16 * DIM;       // [16][HID] 128 KB
  float* gsh = hs + 16 * HID;         // [16] gates
  int*   tsh = (int*)(gsh + 16);      // [16] token ids
  int*   ssh = tsh + 16;              // [16] pair-output slots (-1 = pad row)

  const int t = threadIdx.x;
  if (t < 16) {
    const int base = e * B_TOK + tile * 16;
    const int r    = (t < rows) ? t : 0;         // pad rows replicate row 0
    tsh[t] = list[base + r];
    gsh[t] = (t < rows) ? glist[base + r] : 0.0f;
    ssh[t] = (t < rows) ? rev[base + r] : -1;
  }
  __syncthreads();

#if __has_builtin(__builtin_amdgcn_tensor_load_to_lds) && \
    __has_builtin(__builtin_amdgcn_s_wait_tensorcnt)
  // ---- Tensor Data Mover gather: 16 routed rows of x (4KB each) -> xs ----
  // D# per CDNA5 ISA 8.3/8.4/8.7: gather_mode, 16-bit row indices, 2D tile.
  // This toolchain exposes the 6-arg builtin:
  //   (u32x4 g0, i32x8 g1, i32x4 g2, i32x4 g3, i32x8 g23, i32 cpol)
  if (t < 32) {                       // wave 0 drives the DMA
    const unsigned long long gaddr = (unsigned long long)(uintptr_t)x;
    const unsigned lds_off = (unsigned)(uintptr_t)xs;  // generic-ptr low 32b = LDS offset
    u32x4 g0;
    g0.x = 0x80000001u;               // count=1 | gather_mode=1 | 16-bit indices
    g0.y = (unsigned)__builtin_amdgcn_readfirstlane((int)lds_off);
    g0.z = (unsigned)__builtin_amdgcn_readfirstlane((int)(gaddr & 0xFFFFFFFFull));
    g0.w = ((unsigned)__builtin_amdgcn_readfirstlane(
               (int)((gaddr >> 32) & 0x01FFFFFFull))) | 0x80000000u; // type=2
    i32x8 g1;
    g1[0] = 0x00020000;                               // data_size = 4 bytes
    g1[1] = (int)((unsigned)(DIM & 0xFFFF) << 16);    // tensor_dim0[15:0]
    g1[2] = (int)(((unsigned)(B_TOK & 0xFFFF) << 16) | ((DIM >> 16) & 0xFFFF));
    g1[3] = (int)(((unsigned)(DIM & 0xFFFF) << 16) | ((B_TOK >> 16) & 0xFFFF)); // tile_dim0=DIM
    g1[4] = 16;                                       // tile_dim1 = #valid indices
    g1[5] = DIM;                                      // tensor_dim0_stride[31:0]
    g1[6] = 0;
    g1[7] = 0;
    i32x4 g2, g3;
    i32x8 gidx;                                       // gather row indices (groups 2+3)
    #pragma unroll
    for (int j = 0; j < 4; ++j) {
      const int lo = __builtin_amdgcn_readfirstlane((tsh[2*j]     & 0xFFFF) | (tsh[2*j + 1] << 16));
      const int hi = __builtin_amdgcn_readfirstlane((tsh[8 + 2*j] & 0xFFFF) | (tsh[9 + 2*j] << 16));
      g2[j]       = lo;
      g3[j]       = hi;
      gidx[j]     = lo;
      gidx[4 + j] = hi;
    }
    __builtin_amdgcn_tensor_load_to_lds(g0, g1, g2, g3, gidx, /*cpol=*/0);
    __builtin_amdgcn_s_wait_tensorcnt(0);
  }
#else
  // fallback: cooperative fp32 float4 gather of the 16 routed token rows
  for (int i = t; i < 16 * DIM / 4; i += 256) {
    const int row = i >> 8;
    const int c4  = i & 255;
    ((float4*)(xs + (size_t)row * DIM))[c4] =
        ((const float4*)(x + (size_t)tsh[row] * DIM))[c4];
  }
#endif
  __syncthreads();

  const int wave  = t >> 5;
  const int lane  = t & 31;
  const int row16 = lane & 15;          // A: M index / B,C,D: N index
  const int koff  = (lane >> 4) << 1;   // A/B K pair: lanes 0-15 -> K0/K1, 16-31 -> K2/K3
  const int mbase = (lane >> 4) << 3;   // C/D: lanes 16-31 hold rows 8..15

  // ---- stage 1: hs = relu(xs @ W1[e] + b1[e]) ----
  for (int ct = wave; ct < HID / 16; ct += 8) {
    const int col = ct * 16 + row16;
    v8f c = {0.f, 0.f, 0.f, 0.f, 0.f, 0.f, 0.f, 0.f};
    const float* wp = W1 + ((size_t)e * DIM + koff) * HID + col;
    const float* ap = xs + (size_t)row16 * DIM + koff;
    #pragma unroll 4
    for (int k = 0; k < DIM; k += 4) {
      v2f a = *(const v2f*)(ap + k);
      v2f bvec;
      bvec.x = wp[(size_t)k * HID];
      bvec.y = wp[(size_t)(k + 1) * HID];
      c = wmma_f32(a, bvec, c);
    }
    const float bias = b1[e * HID + col];
    #pragma unroll
    for (int m = 0; m < 8; ++m)
      hs[(size_t)(mbase + m) * HID + col] = fmaxf(c[m] + bias, 0.0f);
  }
  __syncthreads();

  // ---- stage 2: out = hs @ W2[e] + b2[e]; pair_out[slot] = gate * exp(out) ----
  for (int ct = wave; ct < OUT / 16; ct += 8) {
    const int col = ct * 16 + row16;
    v8f c = {0.f, 0.f, 0.f, 0.f, 0.f, 0.f, 0.f, 0.f};
    const float* wp = W2 + ((size_t)e * HID + koff) * OUT + col;
    const float* ap = hs + (size_t)row16 * HID + koff;
    #pragma unroll 4
    for (int k = 0; k < HID; k += 4) {
      v2f a = *(const v2f*)(ap + k);
      v2f bvec;
      bvec.x = wp[(size_t)k * OUT];
      bvec.y = wp[(size_t)(k + 1) * OUT];
      c = wmma_f32(a, bvec, c);
    }
    const float bias = b2[e * OUT + col];
    #pragma unroll
    for (int m = 0; m < 8; ++m) {
      const int r    = mbase + m;
      const int slot = ssh[r];
      if (slot >= 0)
        pair_out[(size_t)slot * OUT + col] = gsh[r] * expf(c[m] + bias);
    }
  }
}

// ---------------------------------------------------------------------------
// 3) y = log(sum of the 4 expert contributions; zeros -> eps)
// ---------------------------------------------------------------------------
__global__ void finalize_kernel(const float* __restrict__ pair_out,
                                float* __restrict__ y, int n) {
  const int i = blockIdx.x * blockDim.x + threadIdx.x;
  if (i < n) {
    const int b = i >> 10;            // OUT == 1024
    const int j = i & (OUT - 1);
    const float* p = pair_out + ((size_t)b * 4) * OUT + j;
    const float s = p[0] + p[OUT] + p[2 * OUT] + p[3 * OUT];
    y[i] = logf(s == 0.0f ? 2.2204460492503131e-16f : s);
  }
}

// ---------------------------------------------------------------------------
// 4) loss = (cv2(importance) + cv2(load)) * 1e-4 , ddof=1
// ---------------------------------------------------------------------------
__global__ void loss_kernel(const float* __restrict__ imp,
                            const float* __restrict__ lod,
                            float* __restrict__ out_loss) {
  if (blockIdx.x == 0 && threadIdx.x == 0) {
    float loss = 0.0f;
    for (int pass = 0; pass < 2; ++pass) {
      const float* v = pass ? lod : imp;
      float mean = 0.0f;
      for (int e = 0; e < NEXP; ++e) mean += v[e];
      mean /= (float)NEXP;
      float var = 0.0f;
      for (int e = 0; e < NEXP; ++e) { const float d = v[e] - mean; var += d * d; }
      var /= (float)(NEXP - 1);
      loss += var / (mean * mean + 1e-10f);
    }
    *out_loss = loss * 1e-4f;
  }
}

// ---------------------------------------------------------------------------
extern "C" void kernel_launch(void* const* d_in, const int* in_sizes, int n_in,
                              void* d_out, int out_size, void* d_ws, size_t ws_size,
                              hipStream_t stream) {
  (void)in_sizes; (void)n_in; (void)out_size; (void)ws_size;
  const float* x     = (const float*)d_in[0];
  const float* wg    = (const float*)d_in[1];
  const float* wn    = (const float*)d_in[2];
  const float* W1    = (const float*)d_in[3];
  const float* b1    = (const float*)d_in[4];
  const float* W2    = (const float*)d_in[5];
  const float* b2    = (const float*)d_in[6];
  const float* noise = (const float*)d_in[7];

  float* y    = (float*)d_out;
  float* loss = y + (size_t)B_TOK * OUT;   // tuple (y, loss) concatenated

  char*  ws       = (char*)d_ws;
  float* pair_out = (float*)ws;                                  // [B*4][O] f32, 16MB
  size_t off      = (size_t)B_TOK * 4 * OUT * sizeof(float);
  int*   cnt      = (int*)(ws + off);                            // [E]
  float* imp      = (float*)(ws + off + 64);                     // [E]
  float* lod      = (float*)(ws + off + 128);                    // [E]
  int*   list     = (int*)(ws + off + 192);                      // [E*B]
  float* glist    = (float*)(ws + off + 192 + (size_t)NEXP * B_TOK * 4);
  int*   rev      = (int*)(ws + off + 192 + (size_t)NEXP * B_TOK * 8);

  zero_kernel<<<1, 64, 0, stream>>>(cnt, imp, lod);
  gating_kernel<<<B_TOK, 256, 0, stream>>>(x, wg, wn, noise, cnt, imp, lod,
                                           list, glist, rev);
  const size_t smem_bytes = (size_t)(16 * DIM + 16 * HID + 64) * sizeof(float); // ~192 KB < 320 KB LDS
  expert_kernel<<<NEXP * 64, 256, smem_bytes, stream>>>(x, W1, b1, W2, b2, cnt,
                                                        list, glist, rev, pair_out);
  finalize_kernel<<<(B_TOK * OUT + 255) / 256, 256, 0, stream>>>(pair_out, y, B_TOK * OUT);
  loss_kernel<<<1, 32, 0, stream>>>(imp, lod, loss);
}